// SQPLoss_41841571397807
// MI455X (gfx1250) — compile-verified
//
#include <hip/hip_runtime.h>
#include <hip/hip_bf16.h>

#define N_CLASSES 32000
#define KCARD     5
#define N_ITERS   50
#define BLOCK     512
#define NWAVES    (BLOCK / 32)
#define CHUNKS    (N_CLASSES / 4)   // 8000 float4 chunks per row (16B each)

// ---------------- wave32 reductions ----------------
__device__ __forceinline__ float wave_red_sum(float v) {
    #pragma unroll
    for (int o = 16; o > 0; o >>= 1) v += __shfl_xor(v, o, 32);
    return v;
}
__device__ __forceinline__ float wave_red_max(float v) {
    #pragma unroll
    for (int o = 16; o > 0; o >>= 1) v = fmaxf(v, __shfl_xor(v, o, 32));
    return v;
}
__device__ __forceinline__ float wave_red_min(float v) {
    #pragma unroll
    for (int o = 16; o > 0; o >>= 1) v = fminf(v, __shfl_xor(v, o, 32));
    return v;
}

// Block-level reductions (deterministic fixed tree). All threads must call.
__device__ __forceinline__ float block_sum(float v, float* red, int lane, int wid) {
    v = wave_red_sum(v);
    __syncthreads();                       // protect red[] from previous use
    if (lane == 0) red[wid] = v;
    __syncthreads();
    if (wid == 0) {
        float t = (lane < NWAVES) ? red[lane] : 0.0f;
        t = wave_red_sum(t);
        if (lane == 0) red[NWAVES] = t;
    }
    __syncthreads();
    return red[NWAVES];
}
__device__ __forceinline__ float block_max(float v, float* red, int lane, int wid) {
    v = wave_red_max(v);
    __syncthreads();
    if (lane == 0) red[wid] = v;
    __syncthreads();
    if (wid == 0) {
        float t = (lane < NWAVES) ? red[lane] : -3.402823466e38f;
        t = wave_red_max(t);
        if (lane == 0) red[NWAVES] = t;
    }
    __syncthreads();
    return red[NWAVES];
}
__device__ __forceinline__ float block_min(float v, float* red, int lane, int wid) {
    v = wave_red_min(v);
    __syncthreads();
    if (lane == 0) red[wid] = v;
    __syncthreads();
    if (wid == 0) {
        float t = (lane < NWAVES) ? red[lane] : 3.402823466e38f;
        t = wave_red_min(t);
        if (lane == 0) red[NWAVES] = t;
    }
    __syncthreads();
    return red[NWAVES];
}

// ---------------- per-row kernel: one workgroup per row ----------------
__global__ __launch_bounds__(BLOCK) void sqp_row_kernel(
        const float* __restrict__ x, const int* __restrict__ y,
        float* __restrict__ rowloss) {
    // Entire 128 KB row lives in LDS (MI455X: 320 KB per WGP).
    __shared__ float buf[N_CLASSES];
    __shared__ float red[NWAVES + 1];

    const int row  = blockIdx.x;
    const int tid  = threadIdx.x;
    const int lane = tid & 31;
    const int wid  = tid >> 5;
    const float* __restrict__ xrow = x + (size_t)row * N_CLASSES;

    // ---- Stage 1: async-copy the row global -> LDS (CDNA5 async path) ----
    for (int k = tid; k < CHUNKS; k += BLOCK) {
        // low 32 bits of a generic pointer to __shared__ == workgroup LDS offset
        unsigned lds_off = (unsigned)(unsigned long long)(const void*)&buf[k * 4];
        unsigned long long ga = (unsigned long long)(const void*)(xrow + k * 4);
        asm volatile("global_load_async_to_lds_b128 %0, %1, off"
                     :: "v"(lds_off), "v"(ga) : "memory");
    }
    asm volatile("s_wait_asynccnt 0" ::: "memory");
    __syncthreads();

    // ---- Stage 2: sum of squares + row max/min (for norm and lo/hi) ----
    float ss = 0.0f, mx = -3.402823466e38f, mn = 3.402823466e38f;
    for (int k = tid; k < CHUNKS; k += BLOCK) {
        float4 v = *(const float4*)&buf[k * 4];
        ss = fmaf(v.x, v.x, ss); ss = fmaf(v.y, v.y, ss);
        ss = fmaf(v.z, v.z, ss); ss = fmaf(v.w, v.w, ss);
        mx = fmaxf(mx, fmaxf(fmaxf(v.x, v.y), fmaxf(v.z, v.w)));
        mn = fminf(mn, fminf(fminf(v.x, v.y), fminf(v.z, v.w)));
    }
    float ssum = block_sum(ss, red, lane, wid);
    float xmx  = block_max(mx, red, lane, wid);
    float xmn  = block_min(mn, red, lane, wid);

    const float rinv = 1.0f / fmaxf(sqrtf(ssum), 1e-12f);  // z = x * rinv
    float lo = -(xmx * rinv) - 10.0f;   // rinv > 0 so max/min order preserved
    float hi = -(xmn * rinv) + 10.0f;

    // ---- Stage 3: overwrite LDS in place with e_i = exp(-z_i) ----
    // |z_i| <= 1 after L2-normalization, so e in [1/e, e]: no range issues.
    for (int k = tid; k < CHUNKS; k += BLOCK) {
        float4 v = *(const float4*)&buf[k * 4];
        v.x = __expf(-v.x * rinv); v.y = __expf(-v.y * rinv);
        v.z = __expf(-v.z * rinv); v.w = __expf(-v.w * rinv);
        *(float4*)&buf[k * 4] = v;
    }
    __syncthreads();

    // ---- Stage 4: 50 bisection iterations entirely out of LDS ----
    // sigmoid(z + mid) = 1 / (1 + e_i * exp(-mid)) : one fma + one v_rcp_f32.
    for (int it = 0; it < N_ITERS; ++it) {
        const float mid = 0.5f * (lo + hi);
        const float em  = __expf(-mid);
        float part = 0.0f;
        for (int k = tid; k < CHUNKS; k += BLOCK) {
            float4 e = *(const float4*)&buf[k * 4];
            part += __builtin_amdgcn_rcpf(fmaf(e.x, em, 1.0f));
            part += __builtin_amdgcn_rcpf(fmaf(e.y, em, 1.0f));
            part += __builtin_amdgcn_rcpf(fmaf(e.z, em, 1.0f));
            part += __builtin_amdgcn_rcpf(fmaf(e.w, em, 1.0f));
        }
        const float s = block_sum(part, red, lane, wid);  // broadcast: uniform
        if (s < (float)KCARD) lo = mid; else hi = mid;
    }

    // ---- Stage 5: differentiable Newton step (forward value) ----
    const float nu0 = 0.5f * (lo + hi);
    const float em0 = __expf(-nu0);
    float gpart = 0.0f, gppart = 0.0f;
    for (int k = tid; k < CHUNKS; k += BLOCK) {
        float4 e = *(const float4*)&buf[k * 4];
        float s0 = __builtin_amdgcn_rcpf(fmaf(e.x, em0, 1.0f));
        float s1 = __builtin_amdgcn_rcpf(fmaf(e.y, em0, 1.0f));
        float s2 = __builtin_amdgcn_rcpf(fmaf(e.z, em0, 1.0f));
        float s3 = __builtin_amdgcn_rcpf(fmaf(e.w, em0, 1.0f));
        gpart  += s0 + s1 + s2 + s3;
        gppart += fmaf(-s0, s0, s0) + fmaf(-s1, s1, s1)
                + fmaf(-s2, s2, s2) + fmaf(-s3, s3, s3);
    }
    const float g  = block_sum(gpart,  red, lane, wid) - (float)KCARD;
    const float gp = block_sum(gppart, red, lane, wid);
    const float nu = nu0 - g / (gp + 1e-12f);

    // ---- Stage 6: target prob + per-row loss (precise division here) ----
    if (tid == 0) {
        int   yv = y[row];
        float ey = buf[yv];                     // e^{-z_y}
        float pt = 1.0f / (1.0f + ey * __expf(-nu));
        rowloss[row] = -logf(pt + 1e-8f);
    }
}

// ---------------- final deterministic mean over rows ----------------
__global__ __launch_bounds__(256) void mean_kernel(
        const float* __restrict__ rowloss, float* __restrict__ out, int n) {
    __shared__ float red[9];
    float v = 0.0f;
    for (int i = threadIdx.x; i < n; i += 256) v += rowloss[i];
    v = wave_red_sum(v);
    const int lane = threadIdx.x & 31, wid = threadIdx.x >> 5;
    if (lane == 0) red[wid] = v;
    __syncthreads();
    if (wid == 0) {
        float t = (lane < 8) ? red[lane] : 0.0f;
        t = wave_red_sum(t);
        if (lane == 0) out[0] = t / (float)n;
    }
}

extern "C" void kernel_launch(void* const* d_in, const int* in_sizes, int n_in,
                              void* d_out, int out_size, void* d_ws, size_t ws_size,
                              hipStream_t stream) {
    const float* x = (const float*)d_in[0];   // [B, 32000] fp32
    const int*   y = (const int*)d_in[1];     // [B] int32
    const int B = in_sizes[1];                // 2048 rows
    float* rowloss = (float*)d_ws;            // B floats of scratch

    sqp_row_kernel<<<B, BLOCK, 0, stream>>>(x, y, rowloss);
    mean_kernel<<<1, 256, 0, stream>>>(rowloss, (float*)d_out, B);
}